// GCN_22325240004826
// MI455X (gfx1250) — compile-verified
//
#include <hip/hip_runtime.h>
#include <math.h>

typedef __attribute__((ext_vector_type(2))) float v2f;
typedef __attribute__((ext_vector_type(8))) float v8f;

#define F_IN 512
#define H1   64
#define H2   32
#define NCLS 16

// ---------------------------------------------------------------------------
// degree / normalization
// ---------------------------------------------------------------------------
__global__ void k_deg_init(float* __restrict__ deg, int n) {
    int i = blockIdx.x * blockDim.x + threadIdx.x;
    if (i < n) deg[i] = 1.0f;                       // self-loop contributes 1
}

__global__ void k_deg_count(const int* __restrict__ dst, float* deg, int e) {
    int i = blockIdx.x * blockDim.x + threadIdx.x;
    if (i < e) unsafeAtomicAdd(&deg[dst[i]], 1.0f); // -> global_atomic_add_f32
}

__global__ void k_rsqrt_inplace(float* d, int n) {
    int i = blockIdx.x * blockDim.x + threadIdx.x;
    if (i < n) d[i] = rsqrtf(d[i]);                 // deg >= 1 always
}

__global__ void k_norm(const int* __restrict__ src, const int* __restrict__ dst,
                       const float* __restrict__ dinv, float* __restrict__ nrm, int e) {
    int i = blockIdx.x * blockDim.x + threadIdx.x;
    if (i < e) nrm[i] = dinv[src[i]] * dinv[dst[i]];
}

// ---------------------------------------------------------------------------
// fp32 WMMA GEMM:  Out[N,NOUT] = act(In[N,K]) @ W[K,NOUT]
//   act(x) = FUSE ? relu(x + bias[kcol]) : x    (fuses previous layer epilogue)
// One wave per 16-row block; A fragment reused across NOUT/16 column tiles.
// ---------------------------------------------------------------------------
template <int K, int NOUT, bool FUSE>
__global__ void gemm_wmma_f32(const float* __restrict__ A,    // [nrows, K]
                              const float* __restrict__ W,    // [K, NOUT]
                              const float* __restrict__ bias, // [K] (prev-layer bias) or null
                              float* __restrict__ Out,        // [nrows, NOUT]
                              int nrows) {
    constexpr int NT = NOUT / 16;
    const int lane  = threadIdx.x & 31;
    const int wave  = blockIdx.x * (blockDim.x >> 5) + (threadIdx.x >> 5);
    const int row0  = wave * 16;
    if (row0 >= nrows) return;                      // wave-uniform: EXEC stays full

    const int m     = lane & 15;                    // row within tile / column of W
    const int khalf = (lane >> 4) * 2;              // 0 or 2 (upper lanes take K+2,K+3)

    v8f acc[NT];
    #pragma unroll
    for (int t = 0; t < NT; ++t) acc[t] = (v8f)0.0f;

    const float* arow = A + (size_t)(row0 + m) * K + khalf;

    for (int k = 0; k < K; k += 4) {
        // A fragment: lane holds A[m][k+khalf], A[m][k+khalf+1] (contiguous -> b64 load)
        v2f a = *(const v2f*)(arow + k);
        if constexpr (FUSE) {
            v2f bb = *(const v2f*)(bias + k + khalf);  // contiguous pair, L0-resident
            a.x = fmaxf(a.x + bb.x, 0.0f);
            a.y = fmaxf(a.y + bb.y, 0.0f);
        }
        #pragma unroll
        for (int t = 0; t < NT; ++t) {
            // B fragment: VGPR v holds W[k+khalf+v][t*16 + m]
            v2f b;
            b.x = W[(size_t)(k + khalf + 0) * NOUT + t * 16 + m];
            b.y = W[(size_t)(k + khalf + 1) * NOUT + t * 16 + m];
            acc[t] = __builtin_amdgcn_wmma_f32_16x16x4_f32(
                false, a, false, b, (short)0, acc[t], false, false);
        }
    }

    // D layout: lane L, VGPR v -> D[v + 8*(L/16)][L%16]
    const int rbase = row0 + (lane >> 4) * 8;
    #pragma unroll
    for (int t = 0; t < NT; ++t)
        #pragma unroll
        for (int v = 0; v < 8; ++v)
            Out[(size_t)(rbase + v) * NOUT + t * 16 + m] = acc[t][v];
}

// ---------------------------------------------------------------------------
// propagation: agg = dinv^2 * h  (self loop; also initializes agg)
//              agg[dst] += norm[e] * h[src]   (atomic scatter, 4 feats/thread)
// ---------------------------------------------------------------------------
template <int D>
__global__ void k_agg_self(const float* __restrict__ h, const float* __restrict__ dinv,
                           float* __restrict__ agg, int n) {
    constexpr int G = D / 4;
    int idx = blockIdx.x * blockDim.x + threadIdx.x;   // over n*G float4 groups
    if (idx >= n * G) return;
    float dv = dinv[idx / G];
    float s  = dv * dv;
    float4 v = ((const float4*)h)[idx];
    v.x *= s; v.y *= s; v.z *= s; v.w *= s;
    ((float4*)agg)[idx] = v;
}

template <int D>
__global__ void k_edge_scatter(const float* __restrict__ h, const int* __restrict__ src,
                               const int* __restrict__ dst, const float* __restrict__ nrm,
                               float* agg, int e) {
    constexpr int G = D / 4;                           // float4 groups per edge
    long long idx = (long long)blockIdx.x * blockDim.x + threadIdx.x;
    if (idx >= (long long)e * G) return;
    int ed = (int)(idx / G);
    int f4 = (int)(idx & (G - 1)) * 4;                 // G is a power of two
    float  nr = nrm[ed];
    float4 v  = *(const float4*)(h + (size_t)src[ed] * D + f4);  // b128 gather
    float* ap = agg + (size_t)dst[ed] * D + f4;
    unsafeAtomicAdd(ap + 0, v.x * nr);
    unsafeAtomicAdd(ap + 1, v.y * nr);
    unsafeAtomicAdd(ap + 2, v.z * nr);
    unsafeAtomicAdd(ap + 3, v.w * nr);
}

__global__ void k_bias_logsoftmax16(const float* __restrict__ agg, const float* __restrict__ b,
                                    float* __restrict__ out, int n) {
    int i = blockIdx.x * blockDim.x + threadIdx.x;
    if (i >= n) return;
    float z[NCLS];
    float mx = -INFINITY;
    #pragma unroll
    for (int c = 0; c < NCLS; ++c) {
        z[c] = agg[(size_t)i * NCLS + c] + b[c];
        mx = fmaxf(mx, z[c]);
    }
    float s = 0.0f;
    #pragma unroll
    for (int c = 0; c < NCLS; ++c) s += expf(z[c] - mx);
    float lse = mx + logf(s);
    #pragma unroll
    for (int c = 0; c < NCLS; ++c) out[(size_t)i * NCLS + c] = z[c] - lse;
}

// ---------------------------------------------------------------------------
extern "C" void kernel_launch(void* const* d_in, const int* in_sizes, int n_in,
                              void* d_out, int out_size, void* d_ws, size_t ws_size,
                              hipStream_t stream) {
    const float* x    = (const float*)d_in[0];
    const int*   eidx = (const int*)  d_in[1];
    const float* W1   = (const float*)d_in[2];
    const float* b1   = (const float*)d_in[3];
    const float* W2   = (const float*)d_in[4];
    const float* b2   = (const float*)d_in[5];
    const float* W3   = (const float*)d_in[6];
    const float* b3   = (const float*)d_in[7];

    const int n = in_sizes[0] / F_IN;     // 100000
    const int e = in_sizes[1] / 2;        // 1600000
    const int* src = eidx;
    const int* dst = eidx + e;

    // workspace carve-out (~58 MB)
    char* ws = (char*)d_ws;
    size_t o = 0;
    auto carve = [&](size_t bytes) -> void* {
        void* p = ws + o;
        o = (o + bytes + 255) & ~(size_t)255;
        return p;
    };
    float* dinv = (float*)carve((size_t)n * sizeof(float));        // deg -> dinv in place
    float* nrm  = (float*)carve((size_t)e * sizeof(float));
    float* bufA = (float*)carve((size_t)n * H1 * sizeof(float));   // h (GEMM out)
    float* bufB = (float*)carve((size_t)n * H1 * sizeof(float));   // agg (scatter target)
    (void)ws_size;

    const int TB = 256;
    auto blk = [&](long long total) { return (int)((total + TB - 1) / TB); };

    // normalization
    k_deg_init<<<blk(n), TB, 0, stream>>>(dinv, n);
    k_deg_count<<<blk(e), TB, 0, stream>>>(dst, dinv, e);
    k_rsqrt_inplace<<<blk(n), TB, 0, stream>>>(dinv, n);
    k_norm<<<blk(e), TB, 0, stream>>>(src, dst, dinv, nrm, e);

    const int gemmBlocks = ((n + 15) / 16 + 7) / 8;   // 8 waves / 256-thread block

    // ---- layer 1: h1 = X@W1 ; agg1 = propagate(h1) ----
    gemm_wmma_f32<F_IN, H1, false><<<gemmBlocks, 256, 0, stream>>>(x, W1, nullptr, bufA, n);
    k_agg_self<H1><<<blk((long long)n * (H1 / 4)), TB, 0, stream>>>(bufA, dinv, bufB, n);
    k_edge_scatter<H1><<<blk((long long)e * (H1 / 4)), TB, 0, stream>>>(bufA, src, dst, nrm, bufB, e);

    // ---- layer 2: h2 = relu(agg1 + b1)@W2 (bias+relu fused into A load) ----
    gemm_wmma_f32<H1, H2, true><<<gemmBlocks, 256, 0, stream>>>(bufB, W2, b1, bufA, n);
    k_agg_self<H2><<<blk((long long)n * (H2 / 4)), TB, 0, stream>>>(bufA, dinv, bufB, n);
    k_edge_scatter<H2><<<blk((long long)e * (H2 / 4)), TB, 0, stream>>>(bufA, src, dst, nrm, bufB, e);

    // ---- layer 3: h3 = relu(agg2 + b2)@W3 ; propagate ; log_softmax(+b3) ----
    gemm_wmma_f32<H2, NCLS, true><<<gemmBlocks, 256, 0, stream>>>(bufB, W3, b2, bufA, n);
    k_agg_self<NCLS><<<blk((long long)n * (NCLS / 4)), TB, 0, stream>>>(bufA, dinv, bufB, n);
    k_edge_scatter<NCLS><<<blk((long long)e * (NCLS / 4)), TB, 0, stream>>>(bufA, src, dst, nrm, bufB, e);
    k_bias_logsoftmax16<<<blk(n), TB, 0, stream>>>(bufB, b3, (float*)d_out, n);
}